// FldzhyanBellNxN_56358560858272
// MI455X (gfx1250) — compile-verified
//
#include <hip/hip_runtime.h>

// N = 128 photonic mesh: M starts diagonal, then 128 steps of
//   M <- mmi_odd @ diag(exp(i*phi)) @ mmi_even @ M,   final diag phase.
// Columns of M evolve independently -> 8 blocks, each owns a 128x16 column
// slab (ping-pong pair) in LDS. One wave per 16x16 output tile; complex GEMM
// via V_WMMA_F32_16X16X4_F32. The banded MMI A-fragments (4 even + 6 odd
// K-chunks per row tile, exact by construction) live entirely in VGPRs, so
// the 256-half-step serial chain touches only LDS + WMMA + 2 barriers/step.

typedef __attribute__((ext_vector_type(2))) float v2f;
typedef __attribute__((ext_vector_type(4))) float v4f;
typedef __attribute__((ext_vector_type(8))) float v8f;

#define N_MESH 128
#define LDS_S  132   // column-major slab stride (floats); conflict-free for 4*l mod 64 pattern

__device__ __forceinline__ v8f wmma_f32(v2f a, v2f b, v8f c) {
  // f32 WMMA: no A/B negate modifiers; plain MAC form.
  return __builtin_amdgcn_wmma_f32_16x16x4_f32(false, a, false, b, (short)0, c, false, false);
}

// C(16x16 complex tile) += A(register fragments, NCH K-chunks) * B(LDS slab)
template<int NCH>
__device__ __forceinline__ void cmatmul_reg(const v2f* are, const v2f* aim, const v2f* aimn,
                                            const float* __restrict__ sRe,
                                            const float* __restrict__ sIm,
                                            int kbase, int m, int hi,
                                            v8f& cre, v8f& cim)
{
  const float* br0 = sRe + m * LDS_S + 2 * hi;   // column m; half select via 2*hi
  const float* bi0 = sIm + m * LDS_S + 2 * hi;
#pragma unroll
  for (int j = 0; j < NCH; ++j) {
    const int k = 4 * (kbase + j);
    v2f br = *reinterpret_cast<const v2f*>(br0 + k);   // ds b64 (2addr-mergeable)
    v2f bi = *reinterpret_cast<const v2f*>(bi0 + k);
    cre = wmma_f32(are[j],  br, cre);
    cre = wmma_f32(aimn[j], bi, cre);
    cim = wmma_f32(are[j],  bi, cim);
    cim = wmma_f32(aim[j],  br, cim);
  }
}

__device__ __forceinline__ void store_tile(float* sRe, float* sIm,
                                           int w, int m, int hi,
                                           const v8f& cre, const v8f& cim)
{
  float* pr = sRe + m * LDS_S + 16 * w + 8 * hi;   // 16B aligned
  float* pi = sIm + m * LDS_S + 16 * w + 8 * hi;
  v4f r0 = { cre[0], cre[1], cre[2], cre[3] };
  v4f r1 = { cre[4], cre[5], cre[6], cre[7] };
  v4f i0 = { cim[0], cim[1], cim[2], cim[3] };
  v4f i1 = { cim[4], cim[5], cim[6], cim[7] };
  *reinterpret_cast<v4f*>(pr)     = r0;
  *reinterpret_cast<v4f*>(pr + 4) = r1;
  *reinterpret_cast<v4f*>(pi)     = i0;
  *reinterpret_cast<v4f*>(pi + 4) = i1;
}

__global__ __launch_bounds__(256) void mesh_propagate_kernel(
    const float* __restrict__ phis,   // (N+2) x 128 float32
    const float* __restrict__ mmiE,   // 128x128 complex64 (interleaved floats)
    const float* __restrict__ mmiO,   // 128x128 complex64
    float* __restrict__ out)          // 128x128 complex64
{
  __shared__ float sRe0[16 * LDS_S], sIm0[16 * LDS_S];   // ping
  __shared__ float sRe1[16 * LDS_S], sIm1[16 * LDS_S];   // pong

  const int tid  = threadIdx.x;
  const int w    = tid >> 5;          // wave id = output row tile (wave32)
  const int lane = tid & 31;
  const int m    = lane & 15;
  const int hi   = lane >> 4;
  const int cb   = blockIdx.x;        // column tile (16 columns)

  // ---- Preload banded A-fragments into registers (loop-invariant). ----
  // mmi_even pairs (2k,2k+1) stay inside a 16-row tile -> chunks 4w..4w+3.
  // mmi_odd pairs (2k+1,2k+2) spill one element across tile edges -> 6 chunks
  // starting at clamp(4w-1, 0, 26); the clamp-padded edge chunk is exactly 0.
  const int e0 = 4 * w;
  int o0 = 4 * w - 1;
  if (o0 < 0)  o0 = 0;
  if (o0 > 26) o0 = 26;

  const int rowg = 16 * w + m;
  const float* arowE = mmiE + 2 * rowg * N_MESH;
  const float* arowO = mmiO + 2 * rowg * N_MESH;

  v2f eRe[4], eIm[4], eImN[4];
  v2f oRe[6], oIm[6], oImN[6];
#pragma unroll
  for (int j = 0; j < 4; ++j) {
    const int k = 4 * (e0 + j) + 2 * hi;
    v4f a = *reinterpret_cast<const v4f*>(arowE + 2 * k);  // (re,im) x {k, k+1}
    eRe[j]  = {  a[0],  a[2] };
    eIm[j]  = {  a[1],  a[3] };
    eImN[j] = { -a[1], -a[3] };
  }
#pragma unroll
  for (int j = 0; j < 6; ++j) {
    const int k = 4 * (o0 + j) + 2 * hi;
    v4f a = *reinterpret_cast<const v4f*>(arowO + 2 * k);
    oRe[j]  = {  a[0],  a[2] };
    oIm[j]  = {  a[1],  a[3] };
    oImN[j] = { -a[1], -a[3] };
  }

  // ---- M = diag(exp(i*phi_0)), columns 16cb..16cb+15, into ping buffer. ----
  for (int idx = tid; idx < 16 * N_MESH; idx += 256) {
    int c = idx >> 7;                 // 0..15
    int r = idx & 127;
    float re = 0.f, im = 0.f;
    if (r == 16 * cb + c) __sincosf(phis[r], &im, &re);
    sRe0[c * LDS_S + r] = re;
    sIm0[c * LDS_S + r] = im;
  }
  __syncthreads();

  // ---- 127 full steps; last step peeled so it writes straight to global. ----
  for (int i = 0; i < N_MESH - 1; ++i) {
    // Phase coeffs for layer i+1 first, so sin/cos + phi load overlap WMMA.
    const float* phiL = phis + (i + 1) * N_MESH + 16 * w + 8 * hi;
    float pc[8], ps[8];
#pragma unroll
    for (int v = 0; v < 8; ++v) __sincosf(phiL[v], &ps[v], &pc[v]);

    v8f cre = {}, cim = {};
    cmatmul_reg<4>(eRe, eIm, eImN, sRe0, sIm0, e0, m, hi, cre, cim);  // mmi_even @ M
#pragma unroll
    for (int v = 0; v < 8; ++v) {                                     // diag phase
      float re = cre[v], im = cim[v];
      cre[v] = re * pc[v] - im * ps[v];
      cim[v] = re * ps[v] + im * pc[v];
    }
    store_tile(sRe1, sIm1, w, m, hi, cre, cim);
    __syncthreads();

    v8f dre = {}, dim_ = {};
    cmatmul_reg<6>(oRe, oIm, oImN, sRe1, sIm1, o0, m, hi, dre, dim_); // mmi_odd @ M
    store_tile(sRe0, sIm0, w, m, hi, dre, dim_);
    __syncthreads();
  }

  // ---- Peeled final step: even + phase(layer 128), odd, phase(layer 129),
  //      then direct register->global writeout of this wave's tile. ----
  {
    const float* phiL = phis + N_MESH * N_MESH + 16 * w + 8 * hi;     // layer N
    float pc[8], ps[8];
#pragma unroll
    for (int v = 0; v < 8; ++v) __sincosf(phiL[v], &ps[v], &pc[v]);

    v8f cre = {}, cim = {};
    cmatmul_reg<4>(eRe, eIm, eImN, sRe0, sIm0, e0, m, hi, cre, cim);
#pragma unroll
    for (int v = 0; v < 8; ++v) {
      float re = cre[v], im = cim[v];
      cre[v] = re * pc[v] - im * ps[v];
      cim[v] = re * ps[v] + im * pc[v];
    }
    store_tile(sRe1, sIm1, w, m, hi, cre, cim);
    __syncthreads();

    v8f dre = {}, dim_ = {};
    cmatmul_reg<6>(oRe, oIm, oImN, sRe1, sIm1, o0, m, hi, dre, dim_);

    const float* phiF = phis + (N_MESH + 1) * N_MESH + 16 * w + 8 * hi;  // layer N+1
#pragma unroll
    for (int v = 0; v < 8; ++v) {
      float s, c;
      __sincosf(phiF[v], &s, &c);
      float re = dre[v], im = dim_[v];
      const int row = 16 * w + 8 * hi + v;
      const int col = 16 * cb + m;
      v2f o = { re * c - im * s, re * s + im * c };
      *reinterpret_cast<v2f*>(out + 2 * (row * N_MESH + col)) = o;     // b64 store
    }
  }
}

extern "C" void kernel_launch(void* const* d_in, const int* in_sizes, int n_in,
                              void* d_out, int out_size, void* d_ws, size_t ws_size,
                              hipStream_t stream) {
  const float* phis = (const float*)d_in[0];   // (130,128) float32
  const float* mmiE = (const float*)d_in[1];   // (128,128) complex64 -> float pairs
  const float* mmiO = (const float*)d_in[2];   // (128,128) complex64 -> float pairs
  float* out = (float*)d_out;                  // (128,128) complex64 -> float pairs

  mesh_propagate_kernel<<<dim3(N_MESH / 16), dim3(256), 0, stream>>>(phis, mmiE, mmiO, out);
}